// KalmanFilter_4200478015710
// MI455X (gfx1250) — compile-verified
//
#include <hip/hip_runtime.h>

#define G 256
#define T 200
#define S 32
#define M 8

typedef __attribute__((ext_vector_type(2))) float v2f;
typedef __attribute__((ext_vector_type(8))) float v8f;

__launch_bounds__(128, 1)
__global__ void kalman_kernel(const float* __restrict__ input,      // [G,T,M]
                              const float* __restrict__ Fg,         // [S,S]
                              const float* __restrict__ Qg,         // [S,S]
                              const float* __restrict__ Hg,         // [M,S]
                              const float* __restrict__ Rg,         // [M,M]
                              const float* __restrict__ init_mean,  // [G,S]
                              const float* __restrict__ init_cov,   // [G,S,S]
                              float* __restrict__ out_mean,         // [G,T,M]
                              float* __restrict__ out_cov)          // [G,T,M,M]
{
    __shared__ float Fs[S][S + 1];
    __shared__ float Qs[S][S];
    __shared__ float Hs[M][S];
    __shared__ float Rs[M][M];
    __shared__ float Pm[S][S + 1];   // current covariance P (prediction)
    __shared__ float Pu[S][S + 1];   // P_u; reused for Pp (pre-symmetrize)
    __shared__ float FP[S][S + 1];   // F * P_u
    __shared__ float HPs[M][S];      // H * P
    __shared__ float Xs[M][S];       // S^{-1} * HP   (K = Xs^T)
    __shared__ float Sm[M][M + 1];   // innovation cov -> Cholesky (in place)
    __shared__ float invd[M];        // 1 / diag(L)
    __shared__ float mv[S], mu[S], Hm[M], dv[M];

    const int g    = blockIdx.x;
    const int tid  = threadIdx.x;
    const int lane = tid & 31;
    const int wave = tid >> 5;

    // ---- one-time loads ----
    for (int i = tid; i < S * S; i += 128) {
        Fs[i >> 5][i & 31] = Fg[i];
        Qs[i >> 5][i & 31] = Qg[i];
        Pm[i >> 5][i & 31] = init_cov[(size_t)g * S * S + i];
    }
    for (int i = tid; i < M * S; i += 128) Hs[i >> 5][i & 31] = Hg[i];
    if (tid < M * M) Rs[tid >> 3][tid & 7] = Rg[tid];
    if (tid < S) mv[tid] = init_mean[g * S + tid];
    __syncthreads();

    // ---- WMMA tile geometry + loop-invariant F fragments in registers ----
    const int tm = (wave >> 1) * 16;
    const int tn = (wave & 1) * 16;
    const int r  = lane & 15;        // A row / B col within tile
    const int kl = (lane >> 4) * 2;  // per-lane K sub-offset
    v2f fA[8];    // stage-1 A fragments:  F[tm+r][kb], F[tm+r][kb+1]
    v2f fBt[8];   // stage-2 B fragments:  F^T[kb][tn+r] = F[tn+r][kb]
    #pragma unroll
    for (int kc = 0; kc < 8; ++kc) {
        const int kb = kc * 4 + kl;
        fA[kc].x  = Fs[tm + r][kb];
        fA[kc].y  = Fs[tm + r][kb + 1];
        fBt[kc].x = Fs[tn + r][kb];
        fBt[kc].y = Fs[tn + r][kb + 1];
    }

    for (int t = 0; t < T; ++t) {
        // ---- HP = H * P (8x32): 2 outputs per thread ----
        {
            int idx = tid * 2;
            #pragma unroll
            for (int u = 0; u < 2; ++u, ++idx) {
                int i = idx >> 5, j = idx & 31;
                float s = 0.f;
                #pragma unroll
                for (int k = 0; k < S; ++k) s = fmaf(Hs[i][k], Pm[k][j], s);
                HPs[i][j] = s;
            }
        }
        // ---- Hm = H * m ; emit measurement mean ----
        if (tid < M) {
            float s = 0.f;
            #pragma unroll
            for (int k = 0; k < S; ++k) s = fmaf(Hs[tid][k], mv[k], s);
            Hm[tid] = s;
            out_mean[((size_t)g * T + t) * M + tid] = s;
        }
        __syncthreads();

        // ---- Smat = HP*H^T + R  (== measurement covariance output) ----
        if (tid < M * M) {
            int i = tid >> 3, j = tid & 7;
            float s = Rs[i][j];
            #pragma unroll
            for (int k = 0; k < S; ++k) s = fmaf(HPs[i][k], Hs[j][k], s);
            Sm[i][j] = s;
            out_cov[(((size_t)g * T + t) * M + i) * M + j] = s;
        }
        if (t == T - 1) break;
        if (tid < M) {
            float y = input[((size_t)g * T + t) * M + tid];
            // hint next step's observation toward the caches
            __builtin_prefetch(&input[((size_t)g * T + t + 1) * M + tid], 0, 0);
            dv[tid] = y - Hm[tid];
        }
        __syncthreads();

        // ---- Cholesky of Sm (8x8, in-place lower) on thread 0 ----
        if (tid == 0) {
            #pragma unroll
            for (int i = 0; i < M; ++i) {
                float s = Sm[i][i];
                for (int k = 0; k < i; ++k) s -= Sm[i][k] * Sm[i][k];
                float lii = sqrtf(fmaxf(s, 1e-30f));
                Sm[i][i] = lii;
                float inv = 1.f / lii;
                invd[i] = inv;
                for (int rr = i + 1; rr < M; ++rr) {
                    float v = Sm[rr][i];
                    for (int k = 0; k < i; ++k) v -= Sm[rr][k] * Sm[i][k];
                    Sm[rr][i] = v * inv;
                }
            }
        }
        __syncthreads();

        // ---- Solve Sm * X = HP : one RHS column per lane (tid<32) ----
        if (tid < S) {
            float id[M];
            #pragma unroll
            for (int i = 0; i < M; ++i) id[i] = invd[i];
            float z[M];
            #pragma unroll
            for (int i = 0; i < M; ++i) {
                float s = HPs[i][tid];
                for (int k = 0; k < i; ++k) s = fmaf(-Sm[i][k], z[k], s);
                z[i] = s * id[i];
            }
            #pragma unroll
            for (int i = M - 1; i >= 0; --i) {
                float s = z[i];
                for (int k = i + 1; k < M; ++k) s = fmaf(-Sm[k][i], z[k], s);
                z[i] = s * id[i];
            }
            #pragma unroll
            for (int i = 0; i < M; ++i) Xs[i][tid] = z[i];
        }
        __syncthreads();

        // ---- P_u = P - X^T*HP ; m_u = m + X^T*d ----
        #pragma unroll
        for (int u = 0; u < 8; ++u) {
            int idx = u * 128 + tid;
            int si = idx >> 5, k = idx & 31;
            float acc = Pm[si][k];
            #pragma unroll
            for (int i = 0; i < M; ++i) acc = fmaf(-Xs[i][si], HPs[i][k], acc);
            Pu[si][k] = acc;
        }
        if (tid < S) {
            float acc = mv[tid];
            #pragma unroll
            for (int i = 0; i < M; ++i) acc = fmaf(Xs[i][tid], dv[i], acc);
            mu[tid] = acc;
        }
        __syncthreads();

        // ---- WMMA stage 1: FP = F * P_u (A = F from registers) ----
        {
            v8f c = {};
            #pragma unroll
            for (int kc = 0; kc < 8; ++kc) {
                const int kb = kc * 4 + kl;
                v2f b;
                b.x = Pu[kb][tn + r];
                b.y = Pu[kb + 1][tn + r];
                c = __builtin_amdgcn_wmma_f32_16x16x4_f32(
                        false, fA[kc], false, b, (short)0, c, false, false);
            }
            #pragma unroll
            for (int v = 0; v < 8; ++v)
                FP[tm + v + (lane >> 4) * 8][tn + r] = c[v];
        }
        __syncthreads();

        // ---- WMMA stage 2: Pp = FP * F^T (B = F^T from registers) ----
        {
            v8f c = {};
            #pragma unroll
            for (int kc = 0; kc < 8; ++kc) {
                const int kb = kc * 4 + kl;
                v2f a;
                a.x = FP[tm + r][kb];
                a.y = FP[tm + r][kb + 1];
                c = __builtin_amdgcn_wmma_f32_16x16x4_f32(
                        false, a, false, fBt[kc], (short)0, c, false, false);
            }
            #pragma unroll
            for (int v = 0; v < 8; ++v)
                Pu[tm + v + (lane >> 4) * 8][tn + r] = c[v];
        }
        __syncthreads();

        // ---- P = 0.5*(Pp + Pp^T) + Q ; m = F * m_u ----
        #pragma unroll
        for (int u = 0; u < 8; ++u) {
            int idx = u * 128 + tid;
            int i = idx >> 5, j = idx & 31;
            Pm[i][j] = 0.5f * (Pu[i][j] + Pu[j][i]) + Qs[i][j];
        }
        if (tid < S) {
            float acc = 0.f;
            #pragma unroll
            for (int k = 0; k < S; ++k) acc = fmaf(Fs[tid][k], mu[k], acc);
            mv[tid] = acc;
        }
        __syncthreads();
    }
}

extern "C" void kernel_launch(void* const* d_in, const int* in_sizes, int n_in,
                              void* d_out, int out_size, void* d_ws, size_t ws_size,
                              hipStream_t stream) {
    const float* input = (const float*)d_in[0];
    const float* Fg    = (const float*)d_in[1];
    const float* Qg    = (const float*)d_in[2];
    const float* Hg    = (const float*)d_in[3];
    const float* Rg    = (const float*)d_in[4];
    const float* im    = (const float*)d_in[5];
    const float* ic    = (const float*)d_in[6];

    float* out_mean = (float*)d_out;                       // [G,T,M]
    float* out_cov  = out_mean + (size_t)G * T * M;        // [G,T,M,M]

    kalman_kernel<<<G, 128, 0, stream>>>(input, Fg, Qg, Hg, Rg, im, ic,
                                         out_mean, out_cov);
}